// STUStudent_81647328297280
// MI455X (gfx1250) — compile-verified
//
#include <hip/hip_runtime.h>
#include <math.h>

// ---------------- problem constants ----------------
#define OBS_DIMC 128
#define ACT_DIMC 32
#define DMODEL   256
#define NLAYERS  2
#define KFILT    16
#define LSEQ     2048
#define BATCH    8
#define NFFT     4096
#define LOG2N    12
#define ROWS     (BATCH * LSEQ)   // 16384 token rows
#define FROWS    (BATCH * NFFT)   // 32768 frequency rows

typedef __attribute__((ext_vector_type(2))) float v2f;
typedef __attribute__((ext_vector_type(8))) float v8f;

// ---------------- WMMA fp32 16x16x4 fragment helpers ----------------
// A (16x4): lanes 0-15 -> M=lane, K=k0..k0+1 ; lanes 16-31 -> M=lane-16, K=k0+2..k0+3
// B (4x16): lanes 0-15 -> N=lane, K=k0..k0+1 ; lanes 16-31 -> N=lane-16, K=k0+2..k0+3
// C (16x16): VGPR j: lanes 0-15 -> (M=j, N=lane); lanes 16-31 -> (M=j+8, N=lane-16)

__device__ __forceinline__ v2f load_a_frag(const float* Ap, int lda, int kk, int lane) {
    int m  = lane & 15;
    int k2 = kk + ((lane >> 4) << 1);
    v2f a;
    a.x = Ap[m * lda + k2];
    a.y = Ap[m * lda + k2 + 1];
    return a;
}

__device__ __forceinline__ v2f load_b_frag(const float* Bp, int ldb, int kk, int lane) {
    int n  = lane & 15;
    int k2 = kk + ((lane >> 4) << 1);
    v2f b;
    b.x = Bp[k2 * ldb + n];
    b.y = Bp[(k2 + 1) * ldb + n];
    return b;
}

__device__ __forceinline__ v8f wmma4(v2f a, v2f b, v8f c) {
    return __builtin_amdgcn_wmma_f32_16x16x4_f32(false, a, false, b,
                                                 (short)0, c, false, false);
}

// =====================================================================
// Fat guard-free WMMA GEMM: C = act(A @ B + bias)
// Block tile 64x128, 8 waves, each wave owns a 32x32 tile (2x2 fragments).
// All dims must be multiples of (64, 128, 16); Kd multiple of 4 for b128.
// =====================================================================
#define GBM 64
#define GBN 128
#define GBK 16
#define GLDA 20            // 80B row pitch: b128-aligned, 16 distinct banks
#define GLDB 132           // 528B row pitch: b128-aligned

__global__ __launch_bounds__(256)
void stu_gemm_big(const float* __restrict__ A, const float* __restrict__ Bm,
                  const float* __restrict__ bias, float* __restrict__ C,
                  int M, int N, int Kd, int act)
{
    __shared__ __align__(16) float As[GBM * GLDA];
    __shared__ __align__(16) float Bs[GBK * GLDB];

    const int tid  = threadIdx.x;
    const int lane = tid & 31;
    const int wave = tid >> 5;
    const int wm   = wave >> 2;   // 0..1  (32-row band)
    const int wn   = wave & 3;    // 0..3  (32-col band)
    const int rowBase = blockIdx.x * GBM;
    const int colBase = blockIdx.y * GBN;

    // staging coordinates (one f4 of A, two f4 of B per thread)
    const int ar  = tid >> 2;            // 0..63
    const int ac4 = (tid & 3) << 2;      // 0,4,8,12
    const int bc0 = tid >> 5;            // 0..7
    const int bn0 = (tid & 31) << 2;     // 0..124
    const int bc1 = bc0 + 8;             // 8..15

    v8f acc00 = {0,0,0,0,0,0,0,0}, acc01 = acc00, acc10 = acc00, acc11 = acc00;

    for (int k0 = 0; k0 < Kd; k0 += GBK) {
        // ---- batched global loads (all issued before any LDS store) ----
        const float4 av  = *(const float4*)(A  + (size_t)(rowBase + ar) * Kd + k0 + ac4);
        const float4 bv0 = *(const float4*)(Bm + (size_t)(k0 + bc0) * N + colBase + bn0);
        const float4 bv1 = *(const float4*)(Bm + (size_t)(k0 + bc1) * N + colBase + bn0);
        if (k0 + GBK < Kd) {
            __builtin_prefetch(A  + (size_t)(rowBase + ar) * Kd + k0 + GBK + ac4, 0, 1);
            __builtin_prefetch(Bm + (size_t)(k0 + GBK + bc0) * N + colBase + bn0, 0, 1);
        }
        *(float4*)(As + ar * GLDA + ac4)  = av;
        *(float4*)(Bs + bc0 * GLDB + bn0) = bv0;
        *(float4*)(Bs + bc1 * GLDB + bn0) = bv1;
        __syncthreads();

        const float* Ap0 = &As[(wm * 32) * GLDA];
        const float* Ap1 = &As[(wm * 32 + 16) * GLDA];
        const float* Bp0 = &Bs[wn * 32];
        const float* Bp1 = &Bs[wn * 32 + 16];
#pragma unroll
        for (int kk = 0; kk < GBK; kk += 4) {
            v2f a0 = load_a_frag(Ap0, GLDA, kk, lane);
            v2f a1 = load_a_frag(Ap1, GLDA, kk, lane);
            v2f b0 = load_b_frag(Bp0, GLDB, kk, lane);
            v2f b1 = load_b_frag(Bp1, GLDB, kk, lane);
            acc00 = wmma4(a0, b0, acc00);
            acc01 = wmma4(a0, b1, acc01);
            acc10 = wmma4(a1, b0, acc10);
            acc11 = wmma4(a1, b1, acc11);
        }
        __syncthreads();
    }

    const int n  = lane & 15;
    const int mo = (lane >> 4) << 3;
    v8f* accs[4] = { &acc00, &acc01, &acc10, &acc11 };
#pragma unroll
    for (int f = 0; f < 4; ++f) {
        int fm = f >> 1, fn = f & 1;
        int gr0 = rowBase + wm * 32 + fm * 16 + mo;
        int gc  = colBase + wn * 32 + fn * 16 + n;
        float bv = bias[gc];
#pragma unroll
        for (int j = 0; j < 8; ++j) {
            float v = (*accs[f])[j] + bv;
            if (act == 1) v = v / (1.f + __expf(-v));   // SiLU
            C[(size_t)(gr0 + j) * N + gc] = v;
        }
    }
}

// =====================================================================
// Spectral contraction GEMM (guard-free, 64x128 tile, 2x2 frags x {re,im}):
// G[b,w,e] = sum_{k,d} F[k,w] * X[b,w,d] * M[k*D+d, e]   (complex A, real B)
// A_re/A_im tiles synthesized in LDS from X and F; B tile shared by both chains.
// =====================================================================
__global__ __launch_bounds__(256)
void stu_spectral_gemm(const float* __restrict__ Xre, const float* __restrict__ Xim,
                       const float* __restrict__ Fre, const float* __restrict__ Fim,
                       const float* __restrict__ Mm,   // (KFILT*DMODEL, DMODEL)
                       float* __restrict__ Gre, float* __restrict__ Gim)
{
    __shared__ __align__(16) float Ar[GBM * GLDA];
    __shared__ __align__(16) float Ai[GBM * GLDA];
    __shared__ __align__(16) float Bs[GBK * GLDB];

    const int tid  = threadIdx.x;
    const int lane = tid & 31;
    const int wave = tid >> 5;
    const int wm   = wave >> 2;
    const int wn   = wave & 3;
    const int rowBase = blockIdx.x * GBM;   // over FROWS
    const int colBase = blockIdx.y * GBN;   // over DMODEL

    const int ar  = tid >> 2;
    const int ac4 = (tid & 3) << 2;
    const int bc0 = tid >> 5;
    const int bn0 = (tid & 31) << 2;
    const int bc1 = bc0 + 8;

    const int arow = rowBase + ar;             // == b*NFFT + w
    const int aw   = arow & (NFFT - 1);

    v8f aR00 = {0,0,0,0,0,0,0,0}, aR01 = aR00, aR10 = aR00, aR11 = aR00;
    v8f aI00 = aR00, aI01 = aR00, aI10 = aR00, aI11 = aR00;

    for (int k0 = 0; k0 < KFILT * DMODEL; k0 += GBK) {
        const int kf = k0 >> 8;           // filter idx: constant per 16-chunk
        const int d  = (k0 & 255) + ac4;
        // ---- batched global loads ----
        const float4 xr4 = *(const float4*)(Xre + (size_t)arow * DMODEL + d);
        const float4 xi4 = *(const float4*)(Xim + (size_t)arow * DMODEL + d);
        const float  fr  = Fre[kf * NFFT + aw];
        const float  fi  = Fim[kf * NFFT + aw];
        const float4 bv0 = *(const float4*)(Mm + (size_t)(k0 + bc0) * DMODEL + colBase + bn0);
        const float4 bv1 = *(const float4*)(Mm + (size_t)(k0 + bc1) * DMODEL + colBase + bn0);
        if (k0 + GBK < KFILT * DMODEL) {
            __builtin_prefetch(Mm + (size_t)(k0 + GBK + bc0) * DMODEL + colBase + bn0, 0, 1);
        }
        float4 ar4, ai4;
        ar4.x = fr * xr4.x - fi * xi4.x;  ai4.x = fr * xi4.x + fi * xr4.x;
        ar4.y = fr * xr4.y - fi * xi4.y;  ai4.y = fr * xi4.y + fi * xr4.y;
        ar4.z = fr * xr4.z - fi * xi4.z;  ai4.z = fr * xi4.z + fi * xr4.z;
        ar4.w = fr * xr4.w - fi * xi4.w;  ai4.w = fr * xi4.w + fi * xr4.w;
        *(float4*)(Ar + ar * GLDA + ac4)  = ar4;
        *(float4*)(Ai + ar * GLDA + ac4)  = ai4;
        *(float4*)(Bs + bc0 * GLDB + bn0) = bv0;
        *(float4*)(Bs + bc1 * GLDB + bn0) = bv1;
        __syncthreads();

        const float* Arp0 = &Ar[(wm * 32) * GLDA];
        const float* Arp1 = &Ar[(wm * 32 + 16) * GLDA];
        const float* Aip0 = &Ai[(wm * 32) * GLDA];
        const float* Aip1 = &Ai[(wm * 32 + 16) * GLDA];
        const float* Bp0  = &Bs[wn * 32];
        const float* Bp1  = &Bs[wn * 32 + 16];
#pragma unroll
        for (int kk = 0; kk < GBK; kk += 4) {
            v2f b0 = load_b_frag(Bp0, GLDB, kk, lane);
            v2f b1 = load_b_frag(Bp1, GLDB, kk, lane);
            v2f r0 = load_a_frag(Arp0, GLDA, kk, lane);
            v2f r1 = load_a_frag(Arp1, GLDA, kk, lane);
            aR00 = wmma4(r0, b0, aR00);
            aR01 = wmma4(r0, b1, aR01);
            aR10 = wmma4(r1, b0, aR10);
            aR11 = wmma4(r1, b1, aR11);
            v2f i0 = load_a_frag(Aip0, GLDA, kk, lane);
            v2f i1 = load_a_frag(Aip1, GLDA, kk, lane);
            aI00 = wmma4(i0, b0, aI00);
            aI01 = wmma4(i0, b1, aI01);
            aI10 = wmma4(i1, b0, aI10);
            aI11 = wmma4(i1, b1, aI11);
        }
        __syncthreads();
    }

    const int n  = lane & 15;
    const int mo = (lane >> 4) << 3;
    v8f* accR[4] = { &aR00, &aR01, &aR10, &aR11 };
    v8f* accI[4] = { &aI00, &aI01, &aI10, &aI11 };
#pragma unroll
    for (int f = 0; f < 4; ++f) {
        int fm = f >> 1, fn = f & 1;
        int gr0 = rowBase + wm * 32 + fm * 16 + mo;
        int gc  = colBase + wn * 32 + fn * 16 + n;
#pragma unroll
        for (int j = 0; j < 8; ++j) {
            size_t o = (size_t)(gr0 + j) * DMODEL + gc;
            Gre[o] = (*accR[f])[j];
            Gim[o] = (*accI[f])[j];
        }
    }
}

// =====================================================================
// Small guarded WMMA GEMM (32x64 tile) — used only for the N=32 output proj.
// =====================================================================
#define SBM 32
#define SBN 64
#define SBK 16
#define SLDA 20
#define SLDB 68

__global__ __launch_bounds__(256)
void stu_gemm_small(const float* __restrict__ A, const float* __restrict__ Bm,
                    const float* __restrict__ bias, float* __restrict__ C,
                    int M, int N, int Kd)
{
    __shared__ __align__(16) float As[SBM * SLDA];
    __shared__ __align__(16) float Bs[SBK * SLDB];

    const int tid  = threadIdx.x;
    const int lane = tid & 31;
    const int wave = tid >> 5;
    const int wm   = wave >> 2;
    const int wn   = wave & 3;
    const int rowBase = blockIdx.x * SBM;
    const int colBase = blockIdx.y * SBN;

    v8f acc = {0,0,0,0,0,0,0,0};

    const int ar  = tid >> 2;            // 0..63 (only 0..31 used for A)
    const int ac4 = (tid & 3) << 2;
    const int bc  = tid >> 4;            // 0..15
    const int bn4 = (tid & 15) << 2;     // 0..60
    const int bgn = colBase + bn4;

    for (int k0 = 0; k0 < Kd; k0 += SBK) {
        float4 av = make_float4(0.f, 0.f, 0.f, 0.f);
        if (ar < SBM) av = *(const float4*)(A + (size_t)(rowBase + ar) * Kd + k0 + ac4);
        float4 bv = make_float4(0.f, 0.f, 0.f, 0.f);
        if (bgn < N)  bv = *(const float4*)(Bm + (size_t)(k0 + bc) * N + bgn);
        if (ar < SBM) *(float4*)(As + ar * SLDA + ac4) = av;
        *(float4*)(Bs + bc * SLDB + bn4) = bv;
        __syncthreads();

        const float* Ap = &As[(wm * 16) * SLDA];
        const float* Bp = &Bs[wn * 16];
#pragma unroll
        for (int kk = 0; kk < SBK; kk += 4) {
            v2f a = load_a_frag(Ap, SLDA, kk, lane);
            v2f b = load_b_frag(Bp, SLDB, kk, lane);
            acc = wmma4(a, b, acc);
        }
        __syncthreads();
    }

    const int n  = lane & 15;
    const int mo = (lane >> 4) << 3;
#pragma unroll
    for (int j = 0; j < 8; ++j) {
        int gr = rowBase + wm * 16 + mo + j;
        int gc = colBase + wn * 16 + n;
        if (gc < N) C[(size_t)gr * N + gc] = acc[j] + bias[gc];
    }
}

// ---------------- in-LDS radix-2 FFT (4096-pt) ----------------
__device__ void stu_fft_inplace(float2* s, int tid, float dir /* -1 fwd, +1 inv */)
{
    for (int stage = 0; stage < LOG2N; ++stage) {
        const int mh = 1 << stage;
        for (int i = tid; i < (NFFT >> 1); i += 256) {
            int grp = i >> stage;
            int pos = i & (mh - 1);
            int i0  = (grp << (stage + 1)) + pos;
            int i1  = i0 + mh;
            float ang = dir * 6.283185307179586f * (float)pos / (float)(mh << 1);
            float sn, cs;
            __sincosf(ang, &sn, &cs);
            float2 b = s[i1];
            float tr = cs * b.x - sn * b.y;
            float ti = cs * b.y + sn * b.x;
            float2 a = s[i0];
            s[i0] = make_float2(a.x + tr, a.y + ti);
            s[i1] = make_float2(a.x - tr, a.y - ti);
        }
        __syncthreads();
    }
}

// forward FFT of h[b,:,d] (zero-padded to 4096), full spectrum out
__global__ __launch_bounds__(256)
void stu_fft_forward(const float* __restrict__ h,
                     float* __restrict__ Xre, float* __restrict__ Xim)
{
    __shared__ float2 s[NFFT];
    const int tid = threadIdx.x;
    const int b = blockIdx.x >> 8;      // DMODEL = 256
    const int d = blockIdx.x & 255;
    for (int t = tid; t < NFFT; t += 256) {
        float xr = (t < LSEQ) ? h[((size_t)b * LSEQ + t) * DMODEL + d] : 0.f;
        int r = __brev((unsigned)t) >> (32 - LOG2N);
        s[r] = make_float2(xr, 0.f);
    }
    __syncthreads();
    stu_fft_inplace(s, tid, -1.f);
    for (int w = tid; w < NFFT; w += 256) {
        size_t o = ((size_t)(b << LOG2N) + w) * DMODEL + d;
        Xre[o] = s[w].x;
        Xim[o] = s[w].y;
    }
}

// FFT of the K filters (and sign-alternated copies)
__global__ __launch_bounds__(256)
void stu_filter_fft(const float* __restrict__ filt,
                    float* __restrict__ FreP, float* __restrict__ FimP,
                    float* __restrict__ FreN, float* __restrict__ FimN)
{
    __shared__ float2 s[NFFT];
    const int tid = threadIdx.x;
    const int k   = blockIdx.x & (KFILT - 1);
    const int neg = blockIdx.x >> 4;
    for (int t = tid; t < NFFT; t += 256) {
        float v = 0.f;
        if (t < LSEQ) {
            v = filt[k * LSEQ + t];
            if (neg && (t & 1)) v = -v;   // sign = +1 even t, -1 odd t
        }
        int r = __brev((unsigned)t) >> (32 - LOG2N);
        s[r] = make_float2(v, 0.f);
    }
    __syncthreads();
    stu_fft_inplace(s, tid, -1.f);
    float* Fre = neg ? FreN : FreP;
    float* Fim = neg ? FimN : FimP;
    for (int w = tid; w < NFFT; w += 256) {
        Fre[k * NFFT + w] = s[w].x;
        Fim[k * NFFT + w] = s[w].y;
    }
}

// inverse FFT of G[b,:,e]; accumulate real part (first L samples) into hspec
__global__ __launch_bounds__(256)
void stu_ifft_acc(const float* __restrict__ Gre, const float* __restrict__ Gim,
                  float* __restrict__ hspec)
{
    __shared__ float2 s[NFFT];
    const int tid = threadIdx.x;
    const int b = blockIdx.x >> 8;
    const int e = blockIdx.x & 255;
    for (int w = tid; w < NFFT; w += 256) {
        size_t o = ((size_t)(b << LOG2N) + w) * DMODEL + e;
        int r = __brev((unsigned)w) >> (32 - LOG2N);
        s[r] = make_float2(Gre[o], Gim[o]);
    }
    __syncthreads();
    stu_fft_inplace(s, tid, +1.f);
    const float inv = 1.f / (float)NFFT;
    for (int t = tid; t < LSEQ; t += 256) {
        size_t o = ((size_t)b * LSEQ + t) * DMODEL + e;
        hspec[o] += s[t].x * inv;
    }
}

// ---------------- fused residual-add + LayerNorm ----------------
__global__ __launch_bounds__(256)
void stu_ln_add(const float* __restrict__ a, const float* __restrict__ bsum,
                const float* __restrict__ g, const float* __restrict__ bb,
                float* __restrict__ out)
{
    __shared__ float red[256];
    const int row = blockIdx.x;
    const int d   = threadIdx.x;
    const size_t o = (size_t)row * DMODEL + d;
    float v = a[o] + bsum[o];
    red[d] = v;
    __syncthreads();
    for (int sft = 128; sft > 0; sft >>= 1) {
        if (d < sft) red[d] += red[d + sft];
        __syncthreads();
    }
    float mu = red[0] * (1.f / DMODEL);
    __syncthreads();
    float dv = v - mu;
    red[d] = dv * dv;
    __syncthreads();
    for (int sft = 128; sft > 0; sft >>= 1) {
        if (d < sft) red[d] += red[d + sft];
        __syncthreads();
    }
    float var = red[0] * (1.f / DMODEL);
    out[o] = dv * rsqrtf(var + 1e-5f) * g[d] + bb[d];
}

// ---------------- host-side orchestration ----------------
extern "C" void kernel_launch(void* const* d_in, const int* in_sizes, int n_in,
                              void* d_out, int out_size, void* d_ws, size_t ws_size,
                              hipStream_t stream)
{
    const float* obs   = (const float*)d_in[0];
    const float* filt  = (const float*)d_in[1];
    const float* W_in  = (const float*)d_in[2];
    const float* b_in  = (const float*)d_in[3];
    const float* Mpos  = (const float*)d_in[4];
    const float* Mneg  = (const float*)d_in[5];
    const float* MdirW = (const float*)d_in[6];
    const float* Mdirb = (const float*)d_in[7];
    const float* ln1g  = (const float*)d_in[8];
    const float* ln1b  = (const float*)d_in[9];
    const float* W1    = (const float*)d_in[10];
    const float* b1    = (const float*)d_in[11];
    const float* W2    = (const float*)d_in[12];
    const float* b2    = (const float*)d_in[13];
    const float* ln2g  = (const float*)d_in[14];
    const float* ln2b  = (const float*)d_in[15];
    const float* W_out = (const float*)d_in[16];
    const float* b_out = (const float*)d_in[17];

    float* ws = (float*)d_ws;
    size_t off = 0;
    float* h     = ws + off; off += (size_t)ROWS * DMODEL;        // 16 MB
    float* Xre   = ws + off; off += (size_t)FROWS * DMODEL;       // 32 MB
    float* Xim   = ws + off; off += (size_t)FROWS * DMODEL;       // 32 MB
    float* Gre   = ws + off; off += (size_t)FROWS * DMODEL;       // 32 MB
    float* Gim   = ws + off; off += (size_t)FROWS * DMODEL;       // 32 MB
    float* hspec = ws + off; off += (size_t)ROWS * DMODEL;        // 16 MB
    float* mlph  = ws + off; off += (size_t)ROWS * 4 * DMODEL;    // 64 MB
    float* mlpo  = ws + off; off += (size_t)ROWS * DMODEL;        // 16 MB
    float* FreP  = ws + off; off += (size_t)KFILT * NFFT;
    float* FimP  = ws + off; off += (size_t)KFILT * NFFT;
    float* FreN  = ws + off; off += (size_t)KFILT * NFFT;
    float* FimN  = ws + off; off += (size_t)KFILT * NFFT;
    (void)ws_size; (void)in_sizes; (void)n_in; (void)out_size;

    const dim3 blk(256);

    // h = obs @ W_in + b_in              (16384 x 256 x 128)
    stu_gemm_big<<<dim3(ROWS / GBM, DMODEL / GBN), blk, 0, stream>>>(
        obs, W_in, b_in, h, ROWS, DMODEL, OBS_DIMC, 0);

    // filter spectra (pos + sign-alternated neg)
    stu_filter_fft<<<dim3(2 * KFILT), blk, 0, stream>>>(filt, FreP, FimP, FreN, FimN);

    for (int i = 0; i < NLAYERS; ++i) {
        // X = FFT(h) per (b,d)
        stu_fft_forward<<<dim3(BATCH * DMODEL), blk, 0, stream>>>(h, Xre, Xim);

        // hspec = h @ Mdir + b (initializes the accumulator buffer)
        stu_gemm_big<<<dim3(ROWS / GBM, DMODEL / GBN), blk, 0, stream>>>(
            h, MdirW + (size_t)i * DMODEL * DMODEL, Mdirb + i * DMODEL,
            hspec, ROWS, DMODEL, DMODEL, 0);

        // positive-filter spectral branch
        stu_spectral_gemm<<<dim3(FROWS / GBM, DMODEL / GBN), blk, 0, stream>>>(
            Xre, Xim, FreP, FimP,
            Mpos + (size_t)i * KFILT * DMODEL * DMODEL, Gre, Gim);
        stu_ifft_acc<<<dim3(BATCH * DMODEL), blk, 0, stream>>>(Gre, Gim, hspec);

        // negative-filter spectral branch
        stu_spectral_gemm<<<dim3(FROWS / GBM, DMODEL / GBN), blk, 0, stream>>>(
            Xre, Xim, FreN, FimN,
            Mneg + (size_t)i * KFILT * DMODEL * DMODEL, Gre, Gim);
        stu_ifft_acc<<<dim3(BATCH * DMODEL), blk, 0, stream>>>(Gre, Gim, hspec);

        // h = LN(h + hspec)
        stu_ln_add<<<dim3(ROWS), blk, 0, stream>>>(
            h, hspec, ln1g + i * DMODEL, ln1b + i * DMODEL, h);

        // MLP
        stu_gemm_big<<<dim3(ROWS / GBM, (4 * DMODEL) / GBN), blk, 0, stream>>>(
            h, W1 + (size_t)i * DMODEL * 4 * DMODEL, b1 + i * 4 * DMODEL,
            mlph, ROWS, 4 * DMODEL, DMODEL, 1);
        stu_gemm_big<<<dim3(ROWS / GBM, DMODEL / GBN), blk, 0, stream>>>(
            mlph, W2 + (size_t)i * 4 * DMODEL * DMODEL, b2 + i * DMODEL,
            mlpo, ROWS, DMODEL, 4 * DMODEL, 0);

        // h = LN(h + mlp)
        stu_ln_add<<<dim3(ROWS), blk, 0, stream>>>(
            h, mlpo, ln2g + i * DMODEL, ln2b + i * DMODEL, h);
    }

    // out = h @ W_out + b_out   (N = 32: guarded small-tile kernel)
    stu_gemm_small<<<dim3(ROWS / SBM, 1), blk, 0, stream>>>(
        h, W_out, b_out, (float*)d_out, ROWS, ACT_DIMC, DMODEL);
}